// GNN_25391846654495
// MI455X (gfx1250) — compile-verified
//
#include <hip/hip_runtime.h>

namespace {

constexpr int kN  = 65536;    // nodes
constexpr int kE  = 1048576;  // edges
constexpr int kDH = 64;       // hidden dim
constexpr int kMaxK = 128;    // max inner dim (layer 1)

typedef __attribute__((ext_vector_type(2))) float v2f;
typedef __attribute__((ext_vector_type(8))) float v8f;

__global__ void fill_kernel(float* __restrict__ p, float v, int n) {
  int i = blockIdx.x * blockDim.x + threadIdx.x;
  if (i < n) p[i] = v;
}

__global__ void deg_accum_kernel(const int* __restrict__ col,
                                 const float* __restrict__ ew,
                                 float* __restrict__ deg) {
  int e = blockIdx.x * blockDim.x + threadIdx.x;
  if (e < kE) atomicAdd(&deg[col[e]], ew[e]);
}

__global__ void dinv_kernel(float* __restrict__ d) {
  int i = blockIdx.x * blockDim.x + threadIdx.x;
  if (i < kN) {
    float v = d[i];
    d[i] = (v > 0.f) ? rsqrtf(v) : 0.f;
  }
}

__global__ void norm_kernel(const int* __restrict__ row, const int* __restrict__ col,
                            const float* __restrict__ ew, const float* __restrict__ dinv,
                            float* __restrict__ norm) {
  int e = blockIdx.x * blockDim.x + threadIdx.x;
  if (e < kE) norm[e] = dinv[row[e]] * ew[e] * dinv[col[e]];
}

// H[N,64] = A[N,K] @ W[K,64] using V_WMMA_F32_16X16X4_F32.
// Block = 128 threads = 4 waves; one 16x16 output tile per wave.
// grid.x = M-tiles/4, grid.y = 4 (64/16 N-tiles).
// All 4 waves share the same 16-column W slice -> stage it in LDS once
// (padded stride 17 => bank = (17k + lane) mod 64 is distinct per lane).
__global__ void gemm_wmma_kernel(const float* __restrict__ A,
                                 const float* __restrict__ W,
                                 float* __restrict__ H, int K) {
  __shared__ float Ws[kMaxK * 17];

  const int tid  = threadIdx.x;
  const int lane = tid & 31;
  const int wave = tid >> 5;
  const int m0 = (blockIdx.x * 4 + wave) * 16;
  const int n0 = blockIdx.y * 16;
  const int half = lane >> 4;  // 0 -> K+{0,1}, 1 -> K+{2,3}
  const int l = lane & 15;

  // Cooperative stage of W[:, n0:n0+16] (K*16 elements, K*16 % 128 == 0:
  // every thread runs the same trip count -> EXEC stays all-ones).
  for (int idx = tid; idx < K * 16; idx += 128) {
    int kk = idx >> 4;
    int nn = idx & 15;
    Ws[kk * 17 + nn] = W[(size_t)kk * kDH + n0 + nn];
  }
  __syncthreads();

  v8f acc = {};
  const float* arow = A + (size_t)(m0 + l) * K;
  #pragma unroll 4
  for (int k = 0; k < K; k += 4) {
    const int ka = k + 2 * half;
    v2f a, b;
    a.x = arow[ka];
    a.y = arow[ka + 1];
    b.x = Ws[ka * 17 + l];
    b.y = Ws[(ka + 1) * 17 + l];
    acc = __builtin_amdgcn_wmma_f32_16x16x4_f32(
        false, a, false, b, (short)0, acc, false, false);
  }

  // C/D layout: VGPR r -> (M = r + 8*half, N = lane%16)
  float* hrow = H + (size_t)(m0 + half * 8) * kDH + (n0 + l);
  #pragma unroll
  for (int r = 0; r < 8; ++r) hrow[(size_t)r * kDH] = acc[r];
}

// 32 consecutive threads handle one edge; each thread covers 2 dims
// (b64 gather of h, 2x global_atomic_add_f32 into L2-resident agg).
__global__ void scatter64_kernel(const int* __restrict__ row, const int* __restrict__ col,
                                 const float* __restrict__ norm,
                                 const float* __restrict__ h,
                                 float* __restrict__ agg) {
  int t = blockIdx.x * blockDim.x + threadIdx.x;
  int e = t >> 5;
  int d = (t & 31) * 2;
  if (e < kE) {
    float nm = norm[e];
    const float* hp = h + (size_t)row[e] * kDH + d;
    float* ap = agg + (size_t)col[e] * kDH + d;
    atomicAdd(ap + 0, nm * hp[0]);
    atomicAdd(ap + 1, nm * hp[1]);
  }
}

// In-place: agg becomes next-layer features (self-loop term, bias, ReLU).
__global__ void epilogue64_kernel(float* __restrict__ agg, const float* __restrict__ h,
                                  const float* __restrict__ dinv,
                                  const float* __restrict__ bias, int relu) {
  int t = blockIdx.x * blockDim.x + threadIdx.x;
  if (t < kN * kDH) {
    int i = t >> 6;
    int d = t & 63;
    float di = dinv[i];
    float v = agg[t] + di * di * h[t] + bias[d];
    if (relu) v = fmaxf(v, 0.f);
    agg[t] = v;
  }
}

// Final 64 -> 2 projection: WMMA would waste 14/16 columns; scalar dot is right.
__global__ void gemm_out_kernel(const float* __restrict__ feat,
                                const float* __restrict__ W8,
                                float* __restrict__ h2) {
  int i = blockIdx.x * blockDim.x + threadIdx.x;
  if (i < kN) {
    float s0 = 0.f, s1 = 0.f;
    const float* f = feat + (size_t)i * kDH;
    #pragma unroll
    for (int k = 0; k < kDH; ++k) {
      s0 = fmaf(f[k], W8[k * 2 + 0], s0);
      s1 = fmaf(f[k], W8[k * 2 + 1], s1);
    }
    h2[i * 2 + 0] = s0;
    h2[i * 2 + 1] = s1;
  }
}

__global__ void scatter2_kernel(const int* __restrict__ row, const int* __restrict__ col,
                                const float* __restrict__ norm,
                                const float* __restrict__ h2,
                                float* __restrict__ agg2) {
  int e = blockIdx.x * blockDim.x + threadIdx.x;
  if (e < kE) {
    float nm = norm[e];
    const float* hp = h2 + (size_t)row[e] * 2;
    float* ap = agg2 + (size_t)col[e] * 2;
    atomicAdd(ap + 0, nm * hp[0]);
    atomicAdd(ap + 1, nm * hp[1]);
  }
}

__global__ void epilogue_out_kernel(const float* __restrict__ agg2,
                                    const float* __restrict__ h2,
                                    const float* __restrict__ dinv,
                                    const float* __restrict__ b8,
                                    float* __restrict__ out) {
  int t = blockIdx.x * blockDim.x + threadIdx.x;
  if (t < kN * 2) {
    int i = t >> 1;
    int d = t & 1;
    float di = dinv[i];
    out[t] = agg2[t] + di * di * h2[t] + b8[d];
  }
}

}  // namespace

extern "C" void kernel_launch(void* const* d_in, const int* in_sizes, int n_in,
                              void* d_out, int out_size, void* d_ws, size_t ws_size,
                              hipStream_t stream) {
  (void)in_sizes; (void)n_in; (void)out_size; (void)ws_size;

  const float* x  = (const float*)d_in[0];
  const int*   ei = (const int*)d_in[1];
  const float* ew = (const float*)d_in[2];
  const float* W[5] = {(const float*)d_in[3], (const float*)d_in[5], (const float*)d_in[7],
                       (const float*)d_in[9], (const float*)d_in[11]};
  const float* B[5] = {(const float*)d_in[4], (const float*)d_in[6], (const float*)d_in[8],
                       (const float*)d_in[10], (const float*)d_in[12]};
  const float* W8 = (const float*)d_in[13];
  const float* b8 = (const float*)d_in[14];
  float* out = (float*)d_out;

  const int* row = ei;        // edge_index[0]: sources
  const int* col = ei + kE;   // edge_index[1]: targets

  // Workspace layout (floats): dinv[N] | norm[E] | featA[N*64] | featB[N*64] | h[N*64]
  float* ws = (float*)d_ws;
  float* dinv  = ws;  ws += kN;
  float* norm  = ws;  ws += kE;
  float* featA = ws;  ws += (size_t)kN * kDH;
  float* featB = ws;  ws += (size_t)kN * kDH;
  float* hbuf  = ws;

  // --- gcn_norm: deg (self-loop = 1) -> dinv -> per-edge norm ---
  fill_kernel<<<kN / 256, 256, 0, stream>>>(dinv, 1.0f, kN);
  deg_accum_kernel<<<kE / 256, 256, 0, stream>>>(col, ew, dinv);
  dinv_kernel<<<kN / 256, 256, 0, stream>>>(dinv);
  norm_kernel<<<kE / 256, 256, 0, stream>>>(row, col, ew, dinv, norm);

  const dim3 ggrid(kN / (16 * 4), kDH / 16);  // 1024 x 4 blocks of 128 threads
  const int nf = kN * kDH;

  const float* cur = x;
  int curK = 128;
  float* bufs[2] = {featA, featB};
  for (int layer = 0; layer < 5; ++layer) {
    float* agg = bufs[layer & 1];
    gemm_wmma_kernel<<<ggrid, 128, 0, stream>>>(cur, W[layer], hbuf, curK);
    fill_kernel<<<nf / 256, 256, 0, stream>>>(agg, 0.f, nf);
    scatter64_kernel<<<(kE * 32) / 256, 256, 0, stream>>>(row, col, norm, hbuf, agg);
    epilogue64_kernel<<<nf / 256, 256, 0, stream>>>(agg, hbuf, dinv, B[layer], 1);
    cur = agg;
    curK = kDH;
  }

  // --- final layer: 64 -> 2, no ReLU, writes d_out ---
  float* agg2 = bufs[1];  // cur == bufs[0] after 5 layers
  gemm_out_kernel<<<kN / 256, 256, 0, stream>>>(cur, W8, hbuf);
  fill_kernel<<<(kN * 2) / 256, 256, 0, stream>>>(agg2, 0.f, kN * 2);
  scatter2_kernel<<<kE / 256, 256, 0, stream>>>(row, col, norm, hbuf, agg2);
  epilogue_out_kernel<<<(kN * 2) / 256, 256, 0, stream>>>(agg2, hbuf, dinv, b8, out);
}